// SapaFeatureAlignment_22960895164781
// MI455X (gfx1250) — compile-verified
//
#include <hip/hip_runtime.h>
#include <math.h>
#include <stdint.h>

// Problem constants (from reference)
#define BB 4
#define HFD 128
#define WFD 128
#define CFD 256
#define HCD 64
#define WCD 64
#define CCD 256
#define ED 64
#define FD 256

typedef float v2f __attribute__((ext_vector_type(2)));
typedef float v8f __attribute__((ext_vector_type(8)));

__device__ __forceinline__ v8f wmma_f32_16x16x4(v2f a, v2f b, v8f c) {
  // D = A(16x4,f32) * B(4x16,f32) + C(16x16,f32), full fp32 matrix pipe
  return __builtin_amdgcn_wmma_f32_16x16x4_f32(
      /*neg_a=*/false, a, /*neg_b=*/false, b,
      /*c_mod=*/(short)0, c, /*reuse_a=*/false, /*reuse_b=*/false);
}

__device__ __forceinline__ float wave_sum32(float v) {
#pragma unroll
  for (int off = 16; off > 0; off >>= 1) v += __shfl_xor(v, off, 32);
  return v;
}

// CDNA5 async memory->LDS copy (ASYNCcnt-tracked, no VGPR staging).
__device__ __forceinline__ void async_ld_b128(unsigned lds_off,
                                              const void* gaddr) {
  asm volatile("global_load_async_to_lds_b128 %0, %1, off"
               :
               : "v"(lds_off), "v"(gaddr)
               : "memory");
}
__device__ __forceinline__ void wait_asynccnt0() {
  asm volatile("s_wait_asynccnt 0" ::: "memory");
}

#define LDSTRIDE 257  // 256 + 1 pad: A-frag column reads hit distinct banks

// -------------------------------------------------------------------------
// Kernel A: coarse layernorm + gate + WMMA GEMMs for qc (N=64), k (N=64),
// v (N=256). One block = 32 coarse pixels (2 M-tiles), 256 threads = 8
// waves. Each wave owns 3 N-tiles (scalar if/else pointers -> global_load,
// NOT flat) and 6 accumulators (2M x 3N): each K-step loads 2 A-frags from
// LDS + 3 B-frags from L2-resident weights and issues 6 independent WMMAs.
// -------------------------------------------------------------------------
__global__ __launch_bounds__(256) void sapa_coarse_proj(
    const float* __restrict__ coarse,
    const float* __restrict__ ln_g, const float* __restrict__ ln_b,
    const float* __restrict__ w_gate, const float* __restrict__ b_gate,
    const float* __restrict__ w_qc, const float* __restrict__ b_qc,
    const float* __restrict__ w_k, const float* __restrict__ b_k,
    const float* __restrict__ w_v, const float* __restrict__ b_v,
    float* __restrict__ gate_out, float* __restrict__ qc_out,
    float* __restrict__ k_out, float* __restrict__ v_out) {
  __shared__ float X[32 * LDSTRIDE];
  const int tid = threadIdx.x;
  const int lane = tid & 31;
  const int wave = tid >> 5;
  const int pbase = blockIdx.x * 32;

  // ---- LayerNorm (4 rows per wave) + gate logit ----
#pragma unroll
  for (int rr = 0; rr < 4; ++rr) {
    const int row = wave * 4 + rr;
    const int p = pbase + row;
    const float* src = coarse + (size_t)p * CCD;
    float4 u0 = *(const float4*)(src + lane * 4);
    float4 u1 = *(const float4*)(src + 128 + lane * 4);
    float s = u0.x + u0.y + u0.z + u0.w + u1.x + u1.y + u1.z + u1.w;
    float sq = u0.x * u0.x + u0.y * u0.y + u0.z * u0.z + u0.w * u0.w +
               u1.x * u1.x + u1.y * u1.y + u1.z * u1.z + u1.w * u1.w;
    s = wave_sum32(s);
    sq = wave_sum32(sq);
    const float mean = s * (1.0f / 256.0f);
    const float var = sq * (1.0f / 256.0f) - mean * mean;
    const float inv = rsqrtf(var + 1e-5f);
    float gdot = 0.0f;
    const float v0[4] = {u0.x, u0.y, u0.z, u0.w};
    const float v1[4] = {u1.x, u1.y, u1.z, u1.w};
#pragma unroll
    for (int i = 0; i < 4; ++i) {
      int c = lane * 4 + i;
      float xh = (v0[i] - mean) * inv * ln_g[c] + ln_b[c];
      X[row * LDSTRIDE + c] = xh;
      gdot = fmaf(xh, w_gate[c], gdot);
    }
#pragma unroll
    for (int i = 0; i < 4; ++i) {
      int c = 128 + lane * 4 + i;
      float xh = (v1[i] - mean) * inv * ln_g[c] + ln_b[c];
      X[row * LDSTRIDE + c] = xh;
      gdot = fmaf(xh, w_gate[c], gdot);
    }
    gdot = wave_sum32(gdot);
    if (lane == 0)
      gate_out[p] = 1.0f / (1.0f + __expf(-(gdot + b_gate[0])));
  }
  __syncthreads();

  // ---- WMMA: 24 N-tiles over concat [qc(4) | k(4) | v(16)] ----
  const int lrow = lane & 15;  // A: M index, B/D: N index
  const int lhi = lane >> 4;   // half-wave selector
  const int t0 = wave * 3 + 0;
  const int t1 = wave * 3 + 1;
  const int t2 = wave * 3 + 2;

  // Scalar pointer selection (no arrays -> keeps global addrspace inference)
  const float *W0, *W1, *W2, *B0, *B1, *B2;
  float *D0, *D1, *D2;
  int N0, N1, N2, c0, c1, c2;
  if (t0 < 4)      { W0 = w_qc; B0 = b_qc; D0 = qc_out; N0 = ED; c0 = t0 * 16; }
  else if (t0 < 8) { W0 = w_k;  B0 = b_k;  D0 = k_out;  N0 = ED; c0 = (t0 - 4) * 16; }
  else             { W0 = w_v;  B0 = b_v;  D0 = v_out;  N0 = FD; c0 = (t0 - 8) * 16; }
  if (t1 < 4)      { W1 = w_qc; B1 = b_qc; D1 = qc_out; N1 = ED; c1 = t1 * 16; }
  else if (t1 < 8) { W1 = w_k;  B1 = b_k;  D1 = k_out;  N1 = ED; c1 = (t1 - 4) * 16; }
  else             { W1 = w_v;  B1 = b_v;  D1 = v_out;  N1 = FD; c1 = (t1 - 8) * 16; }
  if (t2 < 4)      { W2 = w_qc; B2 = b_qc; D2 = qc_out; N2 = ED; c2 = t2 * 16; }
  else if (t2 < 8) { W2 = w_k;  B2 = b_k;  D2 = k_out;  N2 = ED; c2 = (t2 - 4) * 16; }
  else             { W2 = w_v;  B2 = b_v;  D2 = v_out;  N2 = FD; c2 = (t2 - 8) * 16; }

  const float* p0 = W0 + c0 + lrow;
  const float* p1 = W1 + c1 + lrow;
  const float* p2 = W2 + c2 + lrow;

  v8f a00 = {}, a01 = {}, a02 = {}, a10 = {}, a11 = {}, a12 = {};
  for (int k0 = 0; k0 < CCD; k0 += 4) {
    const int ka = k0 + lhi * 2;
    v2f af0, af1;
    af0.x = X[lrow * LDSTRIDE + ka];
    af0.y = X[lrow * LDSTRIDE + ka + 1];
    af1.x = X[(16 + lrow) * LDSTRIDE + ka];
    af1.y = X[(16 + lrow) * LDSTRIDE + ka + 1];
    v2f bf0, bf1, bf2;
    bf0.x = p0[(size_t)ka * N0];
    bf0.y = p0[(size_t)(ka + 1) * N0];
    bf1.x = p1[(size_t)ka * N1];
    bf1.y = p1[(size_t)(ka + 1) * N1];
    bf2.x = p2[(size_t)ka * N2];
    bf2.y = p2[(size_t)(ka + 1) * N2];
    a00 = wmma_f32_16x16x4(af0, bf0, a00);
    a01 = wmma_f32_16x16x4(af0, bf1, a01);
    a02 = wmma_f32_16x16x4(af0, bf2, a02);
    a10 = wmma_f32_16x16x4(af1, bf0, a10);
    a11 = wmma_f32_16x16x4(af1, bf1, a11);
    a12 = wmma_f32_16x16x4(af1, bf2, a12);
  }

  const float bn0 = B0[c0 + lrow];
  const float bn1 = B1[c1 + lrow];
  const float bn2 = B2[c2 + lrow];
#pragma unroll
  for (int j = 0; j < 8; ++j) {
    const int m = j + lhi * 8;
    D0[(size_t)(pbase + m) * N0 + c0 + lrow] = a00[j] + bn0;
    D0[(size_t)(pbase + 16 + m) * N0 + c0 + lrow] = a10[j] + bn0;
    D1[(size_t)(pbase + m) * N1 + c1 + lrow] = a01[j] + bn1;
    D1[(size_t)(pbase + 16 + m) * N1 + c1 + lrow] = a11[j] + bn1;
    D2[(size_t)(pbase + m) * N2 + c2 + lrow] = a02[j] + bn2;
    D2[(size_t)(pbase + 16 + m) * N2 + c2 + lrow] = a12[j] + bn2;
  }
}

// -------------------------------------------------------------------------
// Kernel B: fine layernorm + WMMA qf GEMM + gate blend with upsampled qc.
// One block = 64 fine pixels (4 M-tiles), 8 waves: wave = (mpair, ntile),
// 2 accumulators per wave. Dynamic LDS (64*257 floats ~= 65.8 KB).
// -------------------------------------------------------------------------
__global__ __launch_bounds__(256) void sapa_fine_query(
    const float* __restrict__ fine,
    const float* __restrict__ ln_g, const float* __restrict__ ln_b,
    const float* __restrict__ w_qf, const float* __restrict__ b_qf,
    const float* __restrict__ gate_in, const float* __restrict__ qc_in,
    float* __restrict__ query_out) {
  extern __shared__ float X[];
  const int tid = threadIdx.x;
  const int lane = tid & 31;
  const int wave = tid >> 5;
  const int pbase = blockIdx.x * 64;

#pragma unroll
  for (int rr = 0; rr < 8; ++rr) {
    const int row = wave * 8 + rr;
    const int p = pbase + row;
    const float* src = fine + (size_t)p * CFD;
    float4 u0 = *(const float4*)(src + lane * 4);
    float4 u1 = *(const float4*)(src + 128 + lane * 4);
    float s = u0.x + u0.y + u0.z + u0.w + u1.x + u1.y + u1.z + u1.w;
    float sq = u0.x * u0.x + u0.y * u0.y + u0.z * u0.z + u0.w * u0.w +
               u1.x * u1.x + u1.y * u1.y + u1.z * u1.z + u1.w * u1.w;
    s = wave_sum32(s);
    sq = wave_sum32(sq);
    const float mean = s * (1.0f / 256.0f);
    const float var = sq * (1.0f / 256.0f) - mean * mean;
    const float inv = rsqrtf(var + 1e-5f);
    const float v0[4] = {u0.x, u0.y, u0.z, u0.w};
    const float v1[4] = {u1.x, u1.y, u1.z, u1.w};
#pragma unroll
    for (int i = 0; i < 4; ++i) {
      int c = lane * 4 + i;
      X[row * LDSTRIDE + c] = (v0[i] - mean) * inv * ln_g[c] + ln_b[c];
    }
#pragma unroll
    for (int i = 0; i < 4; ++i) {
      int c = 128 + lane * 4 + i;
      X[row * LDSTRIDE + c] = (v1[i] - mean) * inv * ln_g[c] + ln_b[c];
    }
  }
  __syncthreads();

  const int lrow = lane & 15;
  const int lhi = lane >> 4;
  const int ntile = wave & 3;
  const int mbase = (wave >> 2) * 32;  // rows mbase..mbase+31 (2 M-tiles)
  const float* bp = w_qf + ntile * 16 + lrow;
  v8f acc0 = {}, acc1 = {};
  for (int k0 = 0; k0 < CFD; k0 += 4) {
    const int ka = k0 + lhi * 2;
    v2f af0, af1, bfrag;
    af0.x = X[(mbase + lrow) * LDSTRIDE + ka];
    af0.y = X[(mbase + lrow) * LDSTRIDE + ka + 1];
    af1.x = X[(mbase + 16 + lrow) * LDSTRIDE + ka];
    af1.y = X[(mbase + 16 + lrow) * LDSTRIDE + ka + 1];
    bfrag.x = bp[(size_t)ka * ED];
    bfrag.y = bp[(size_t)(ka + 1) * ED];
    acc0 = wmma_f32_16x16x4(af0, bfrag, acc0);
    acc1 = wmma_f32_16x16x4(af1, bfrag, acc1);
  }
  const int n = ntile * 16 + lrow;
  const float bn = b_qf[n];
#pragma unroll
  for (int mt = 0; mt < 2; ++mt) {
    const v8f acc = (mt == 0) ? acc0 : acc1;
#pragma unroll
    for (int j = 0; j < 8; ++j) {
      const int m = mbase + mt * 16 + j + lhi * 8;
      const int p = pbase + m;
      const int bidx = p >> 14;
      const int rem = p & 16383;
      const int h = rem >> 7;
      const int w = rem & 127;
      const int cp = (bidx * HCD + (h >> 1)) * WCD + (w >> 1);
      const float g = gate_in[cp];
      const float q =
          (acc[j] + bn) * g + qc_in[(size_t)cp * ED + n] * (1.0f - g);
      query_out[(size_t)p * ED + n] = q;
    }
  }
}

// -------------------------------------------------------------------------
// Kernel C: 5x5 local attention. One block = 4x4 coarse cells (64 fine px).
// 8x8 zero-padded k/v halo + 64-pixel query tile staged into LDS with
// CDNA5 GLOBAL_LOAD_ASYNC_TO_LDS_B128 (ASYNCcnt). Dynamic LDS ~102.5 KB.
// -------------------------------------------------------------------------
__global__ __launch_bounds__(256) void sapa_attention(
    const float* __restrict__ query, const float* __restrict__ kglb,
    const float* __restrict__ vglb, float* __restrict__ out) {
  extern __shared__ float smem[];
  float* kbuf = smem;                    // 8*8*64   = 4096 floats
  float* vbuf = kbuf + 8 * 8 * 64;       // 8*8*256  = 16384 floats
  float* qbuf = vbuf + 8 * 8 * 256;      // 64*64    = 4096 floats
  float* attb = qbuf + 64 * 64;          // 64*26    = 1664 floats

  const int tid = threadIdx.x;
  const int b = blockIdx.z;
  const int h0 = blockIdx.y * 4;
  const int w0 = blockIdx.x * 4;

  // ---- cooperative staging: k/v halo (zero-pad OOB) + query tile ----
  {
    const int cell = tid >> 2;  // 0..63
    const int part = tid & 3;
    const int ly = cell >> 3, lx = cell & 7;
    const int hh = h0 - 2 + ly, ww = w0 - 2 + lx;
    const bool ok = (hh >= 0) && (hh < HCD) && (ww >= 0) && (ww < WCD);
    const size_t cgi = (size_t)(b * HCD + (ok ? hh : 0)) * WCD + (ok ? ww : 0);
    const float* ks = kglb + cgi * ED + part * 16;
    const float* vs = vglb + cgi * FD + part * 64;
    float* kd = kbuf + cell * ED + part * 16;
    float* vd = vbuf + cell * FD + part * 64;
    if (ok) {
#pragma unroll
      for (int i = 0; i < 4; ++i)
        async_ld_b128((unsigned)(uintptr_t)(kd + i * 4), ks + i * 4);
#pragma unroll
      for (int i = 0; i < 16; ++i)
        async_ld_b128((unsigned)(uintptr_t)(vd + i * 4), vs + i * 4);
    } else {
      const float4 z = make_float4(0.f, 0.f, 0.f, 0.f);
#pragma unroll
      for (int i = 0; i < 4; ++i) ((float4*)kd)[i] = z;
#pragma unroll
      for (int i = 0; i < 16; ++i) ((float4*)vd)[i] = z;
    }
    // query tile: pixel = cell with same (cy,cx,sy,sx) decode as phases 1/3
    const int cy = cell >> 4, cx = (cell >> 2) & 3;
    const int sy = (cell >> 1) & 1, sx = cell & 1;
    const int fh = (h0 + cy) * 2 + sy;
    const int fw = (w0 + cx) * 2 + sx;
    const float* qs =
        query + ((size_t)(b * HFD + fh) * WFD + fw) * ED + part * 16;
    float* qd = qbuf + cell * ED + part * 16;
#pragma unroll
    for (int i = 0; i < 4; ++i)
      async_ld_b128((unsigned)(uintptr_t)(qd + i * 4), qs + i * 4);
  }
  wait_asynccnt0();
  __syncthreads();

  // ---- phase 1: attention logits (25 per pixel, split over 4 groups) ----
  {
    const int pid = tid & 63;
    const int grp = tid >> 6;
    const int cy = pid >> 4, cx = (pid >> 2) & 3;
    const float4* qp = (const float4*)(qbuf + pid * ED);
    float4 qv[16];
#pragma unroll
    for (int i = 0; i < 16; ++i) qv[i] = qp[i];
    for (int p = grp; p < 25; p += 4) {
      const int py = p / 5, px = p - py * 5;
      const float4* kp =
          (const float4*)(kbuf + ((cy + py) * 8 + (cx + px)) * ED);
      float dot = 0.0f;
#pragma unroll
      for (int i = 0; i < 16; ++i) {
        const float4 kv = kp[i];
        dot = fmaf(qv[i].x, kv.x, dot);
        dot = fmaf(qv[i].y, kv.y, dot);
        dot = fmaf(qv[i].z, kv.z, dot);
        dot = fmaf(qv[i].w, kv.w, dot);
      }
      attb[pid * 26 + p] = dot;
    }
  }
  __syncthreads();

  // ---- phase 2: softmax over 25 ----
  if (tid < 64) {
    float* a = attb + tid * 26;
    float mx = a[0];
#pragma unroll
    for (int p = 1; p < 25; ++p) mx = fmaxf(mx, a[p]);
    float s = 0.0f;
#pragma unroll
    for (int p = 0; p < 25; ++p) {
      const float e = __expf(a[p] - mx);
      a[p] = e;
      s += e;
    }
    const float inv = 1.0f / s;
#pragma unroll
    for (int p = 0; p < 25; ++p) a[p] *= inv;
  }
  __syncthreads();

  // ---- phase 3: weighted value accumulation (4 threads x 64 ch / pixel) --
  {
    const int pid = tid >> 2;
    const int c0 = (tid & 3) * 64;
    const int cy = pid >> 4, cx = (pid >> 2) & 3;
    const int sy = (pid >> 1) & 1, sx = pid & 1;
    const int fh = (h0 + cy) * 2 + sy;
    const int fw = (w0 + cx) * 2 + sx;
    float4 acc[16];
#pragma unroll
    for (int i = 0; i < 16; ++i) acc[i] = make_float4(0.f, 0.f, 0.f, 0.f);
    for (int p = 0; p < 25; ++p) {
      const float w = attb[pid * 26 + p];
      const int py = p / 5, px = p - py * 5;
      const float4* vp =
          (const float4*)(vbuf + ((cy + py) * 8 + (cx + px)) * FD + c0);
#pragma unroll
      for (int i = 0; i < 16; ++i) {
        const float4 vv = vp[i];
        acc[i].x = fmaf(w, vv.x, acc[i].x);
        acc[i].y = fmaf(w, vv.y, acc[i].y);
        acc[i].z = fmaf(w, vv.z, acc[i].z);
        acc[i].w = fmaf(w, vv.w, acc[i].w);
      }
    }
    float4* op = (float4*)(out + ((size_t)(b * HFD + fh) * WFD + fw) * FD + c0);
#pragma unroll
    for (int i = 0; i < 16; ++i) op[i] = acc[i];
  }
}

// -------------------------------------------------------------------------
extern "C" void kernel_launch(void* const* d_in, const int* in_sizes, int n_in,
                              void* d_out, int out_size, void* d_ws,
                              size_t ws_size, hipStream_t stream) {
  const float* fine = (const float*)d_in[0];
  const float* coarse = (const float*)d_in[1];
  const float* ln_f_g = (const float*)d_in[2];
  const float* ln_f_b = (const float*)d_in[3];
  const float* ln_c_g = (const float*)d_in[4];
  const float* ln_c_b = (const float*)d_in[5];
  const float* w_gate = (const float*)d_in[6];
  const float* b_gate = (const float*)d_in[7];
  const float* w_qf = (const float*)d_in[8];
  const float* b_qf = (const float*)d_in[9];
  const float* w_qc = (const float*)d_in[10];
  const float* b_qc = (const float*)d_in[11];
  const float* w_k = (const float*)d_in[12];
  const float* b_k = (const float*)d_in[13];
  const float* w_v = (const float*)d_in[14];
  const float* b_v = (const float*)d_in[15];
  float* out = (float*)d_out;

  // Workspace layout (floats): gate | qc | k | v | query  (~42 MB total)
  float* ws = (float*)d_ws;
  float* gate = ws;                                 // 16384
  float* qc = gate + (size_t)BB * HCD * WCD;        // 1048576
  float* kk = qc + (size_t)BB * HCD * WCD * ED;     // 1048576
  float* vv = kk + (size_t)BB * HCD * WCD * ED;     // 4194304
  float* query = vv + (size_t)BB * HCD * WCD * FD;  // 4194304

  const int coarsePixels = BB * HCD * WCD;  // 16384
  const int finePixels = BB * HFD * WFD;    // 65536

  sapa_coarse_proj<<<coarsePixels / 32, 256, 0, stream>>>(
      coarse, ln_c_g, ln_c_b, w_gate, b_gate, w_qc, b_qc, w_k, b_k, w_v, b_v,
      gate, qc, kk, vv);

  const size_t fineq_lds = (size_t)64 * LDSTRIDE * sizeof(float);
  sapa_fine_query<<<finePixels / 64, 256, fineq_lds, stream>>>(
      fine, ln_f_g, ln_f_b, w_qf, b_qf, gate, qc, query);

  const size_t attn_lds =
      (8 * 8 * 64 + 8 * 8 * 256 + 64 * 64 + 64 * 26) * sizeof(float);
  sapa_attention<<<dim3(WCD / 4, HCD / 4, BB), 256, attn_lds, stream>>>(
      query, kk, vv, out);
}